// Net_12730283065591
// MI455X (gfx1250) — compile-verified
//
#include <hip/hip_runtime.h>
#include <hip/hip_bf16.h>
#include <math.h>

// Problem dims (compile-time)
#define N_IN 3
#define NH1 4
#define NH2 8
#define NL 4

// LDS constant-block layout (dword offsets)
#define OW0 0    // 12
#define OB0 12   // 4
#define OW1 16   // 32
#define OB1 48   // 8
#define OW2 56   // 32
#define OB2 88   // 4
#define OLM 92   // 3
#define CTOT 95

typedef __attribute__((ext_vector_type(4))) unsigned int u32x4;
typedef __attribute__((ext_vector_type(8))) int i32x8;
typedef __attribute__((ext_vector_type(4))) int i32x4;

__device__ __forceinline__ float fast_tanh(float x) {
#if __has_builtin(__builtin_amdgcn_tanhf)
  return __builtin_amdgcn_tanhf(x);            // v_tanh_f32 (gfx1250 transcendental)
#elif __has_builtin(__builtin_amdgcn_tanh_f32)
  return __builtin_amdgcn_tanh_f32(x);
#else
  return tanhf(x);
#endif
}

#if defined(__gfx1250__) && __has_builtin(__builtin_amdgcn_tensor_load_to_lds) && __has_builtin(__builtin_amdgcn_s_wait_tensorcnt)
#define PINN_USE_TDM 1
#else
#define PINN_USE_TDM 0
#endif

#if PINN_USE_TDM
// One 1-D tensor (ndw dwords) Global -> LDS via the Tensor Data Mover.
// D# built per CDNA5 ISA 8.3/8.4: group0 {count=1, lds_addr, global_addr, type=2},
// group1 {data_size=4B, tensor_dim0=ndw, tensor_dim1=1, tile_dim0=ndw, tile_dim1=1,
// stride0=ndw}. Groups 2/3 zero (<=2D tensor). Tracked by TENSORcnt.
__device__ __forceinline__ void tdm_load_1d(const float* gsrc, unsigned lds_off, unsigned ndw) {
  unsigned long long ga = (unsigned long long)(uintptr_t)gsrc;
  u32x4 g0;
  g0.x = 1u;                                              // count=1, user mode, no gather
  g0.y = lds_off;                                         // lds_addr (bytes)
  g0.z = (unsigned)(ga & 0xffffffffull);                  // global_addr[31:0]
  g0.w = (unsigned)((ga >> 32) & 0x01ffffffull) | (2u << 30); // addr[56:32] | type=2
  i32x8 g1;
  g1[0] = (int)(2u << 16);                                // wg_mask=0, data_size=2 (4 bytes)
  g1[1] = (int)((ndw & 0xffffu) << 16);                   // tensor_dim0[15:0]
  g1[2] = (int)(((ndw >> 16) & 0xffffu) | (1u << 16));    // tensor_dim0[31:16] | tensor_dim1=1
  g1[3] = (int)((ndw & 0xffffu) << 16);                   // tensor_dim1 hi=0 | tile_dim0=ndw
  g1[4] = 1;                                              // tile_dim1=1, tile_dim2=0
  g1[5] = (int)ndw;                                       // tensor_dim0_stride lo
  g1[6] = 0;                                              // stride0 hi | stride1 lo
  g1[7] = 0;                                              // stride1 hi
  i32x4 z2 = {0, 0, 0, 0};
  i32x4 z3 = {0, 0, 0, 0};
#if defined(__clang_major__) && (__clang_major__ >= 23)
  // amdgpu-toolchain (clang-23 / therock-10.0 headers): 6-arg builtin
  i32x8 z4 = {0, 0, 0, 0, 0, 0, 0, 0};
  __builtin_amdgcn_tensor_load_to_lds(g0, g1, z2, z3, z4, 0);
#else
  // ROCm 7.2 (clang-22): 5-arg builtin
  __builtin_amdgcn_tensor_load_to_lds(g0, g1, z2, z3, 0);
#endif
}
#endif

__global__ __launch_bounds__(256) void pinn_fused_kernel(
    const float* __restrict__ x,
    const float* __restrict__ W0, const float* __restrict__ b0,
    const float* __restrict__ W1, const float* __restrict__ b1,
    const float* __restrict__ W2, const float* __restrict__ b2,
    const float* __restrict__ lmb,
    float* __restrict__ out, int B)
{
  __shared__ __align__(16) float sc[96];

#if PINN_USE_TDM
  if (threadIdx.x == 0) {
    unsigned base = (unsigned)(uintptr_t)&sc[0];          // low 32 bits = LDS byte offset
    tdm_load_1d(W0,  base + OW0 * 4, 12);
    tdm_load_1d(b0,  base + OB0 * 4, 4);
    tdm_load_1d(W1,  base + OW1 * 4, 32);
    tdm_load_1d(b1,  base + OB1 * 4, 8);
    tdm_load_1d(W2,  base + OW2 * 4, 32);
    tdm_load_1d(b2,  base + OB2 * 4, 4);
    tdm_load_1d(lmb, base + OLM * 4, 3);
    __builtin_amdgcn_s_wait_tensorcnt(0);                 // TENSORcnt==0 -> LDS data valid
  }
  __syncthreads();
#else
  {
    int t = (int)threadIdx.x;
    if (t < CTOT) {
      float v;
      if      (t < 12) v = W0[t];
      else if (t < 16) v = b0[t - 12];
      else if (t < 48) v = W1[t - 16];
      else if (t < 56) v = b1[t - 48];
      else if (t < 88) v = W2[t - 56];
      else if (t < 92) v = b2[t - 88];
      else             v = lmb[t - 92];
      sc[t] = v;
    }
  }
  __syncthreads();
#endif

  // ---- hoist constants LDS -> registers (broadcast ds_loads, once per thread) ----
  float W0r[NH1][N_IN], b0r[NH1], W1r[NH2][NH1], b1r[NH2], W2r[NL][NH2], b2r[NL], lmr[NL - 1];
  float g0c[NH1];
#pragma unroll
  for (int m = 0; m < NH1; ++m) {
#pragma unroll
    for (int a = 0; a < N_IN; ++a) W0r[m][a] = sc[OW0 + m * N_IN + a];
    b0r[m] = sc[OB0 + m];
    g0c[m] = W0r[m][0] * W0r[m][0] + W0r[m][1] * W0r[m][1] - W0r[m][2] * W0r[m][2];
  }
#pragma unroll
  for (int p = 0; p < NH2; ++p) {
#pragma unroll
    for (int m = 0; m < NH1; ++m) W1r[p][m] = sc[OW1 + p * NH1 + m];
    b1r[p] = sc[OB1 + p];
  }
#pragma unroll
  for (int j = 0; j < NL; ++j) {
#pragma unroll
    for (int p = 0; p < NH2; ++p) W2r[j][p] = sc[OW2 + j * NH2 + p];
    b2r[j] = sc[OB2 + j];
  }
#pragma unroll
  for (int p = 0; p < NL - 1; ++p) lmr[p] = sc[OLM + p];

  const int stride = (int)(gridDim.x * blockDim.x);
  for (int i = (int)(blockIdx.x * blockDim.x + threadIdx.x); i < B; i += stride) {
    const size_t jx = 3ull * (unsigned)i;
    const float xv0 = x[jx + 0], xv1 = x[jx + 1], xv2 = x[jx + 2];

    // ---- forward (also the derivative-path activations; values identical) ----
    float h0[NH1], d0[NH1], s0[NH1];
#pragma unroll
    for (int m = 0; m < NH1; ++m) {
      float z = fmaf(W0r[m][0], xv0, fmaf(W0r[m][1], xv1, fmaf(W0r[m][2], xv2, b0r[m])));
      float h = fast_tanh(z);
      h0[m] = h;
      d0[m] = fmaf(-h, h, 1.0f);
      s0[m] = -2.0f * h * d0[m];
    }
    float h1[NH2], d1[NH2], s1[NH2];
#pragma unroll
    for (int p = 0; p < NH2; ++p) {
      float z = b1r[p];
#pragma unroll
      for (int m = 0; m < NH1; ++m) z = fmaf(W1r[p][m], h0[m], z);
      float h = fast_tanh(z);
      h1[p] = h;
      d1[p] = fmaf(-h, h, 1.0f);
      s1[p] = -2.0f * h * d1[p];
    }
    float h2[NL], d2[NL], s2[NL];
#pragma unroll
    for (int j = 0; j < NL; ++j) {
      float z = b2r[j];
#pragma unroll
      for (int p = 0; p < NH2; ++p) z = fmaf(W2r[j][p], h1[p], z);
      float h = fast_tanh(z);
      h2[j] = h;
      d2[j] = fmaf(-h, h, 1.0f);
      s2[j] = -2.0f * h * d2[j];
    }

    // ---- FW0 = diag(d0) W0 ; Q1 = W1 FW0 ; g1, e1 ----
    float FW0[NH1][N_IN];
#pragma unroll
    for (int m = 0; m < NH1; ++m)
#pragma unroll
      for (int a = 0; a < N_IN; ++a) FW0[m][a] = d0[m] * W0r[m][a];

    float Q1[NH2][N_IN], e1[NH2];
#pragma unroll
    for (int p = 0; p < NH2; ++p) {
      float q0 = 0.f, q1 = 0.f, q2 = 0.f;
#pragma unroll
      for (int m = 0; m < NH1; ++m) {
        q0 = fmaf(W1r[p][m], FW0[m][0], q0);
        q1 = fmaf(W1r[p][m], FW0[m][1], q1);
        q2 = fmaf(W1r[p][m], FW0[m][2], q2);
      }
      Q1[p][0] = q0; Q1[p][1] = q1; Q1[p][2] = q2;
      float g1 = fmaf(q1, q1, fmaf(q0, q0, -(q2 * q2)));
      e1[p] = s1[p] * g1;
    }

    // ---- e0[m] = s0 g0 - d0 W0[:,2] ; v[p] = d1 (W1 e0) + e1 ----
    float e0[NH1];
#pragma unroll
    for (int m = 0; m < NH1; ++m) e0[m] = fmaf(s0[m], g0c[m], -(d0[m] * W0r[m][2]));

    float vt[NH2];
#pragma unroll
    for (int p = 0; p < NH2; ++p) {
      float t = 0.f;
#pragma unroll
      for (int m = 0; m < NH1; ++m) t = fmaf(W1r[p][m], e0[m], t);
      vt[p] = fmaf(d1[p], t, e1[p]);
    }

    // ---- Q2 = W2 diag(d1) Q1 ; g2 ----
    float Rm[NH2][N_IN];
#pragma unroll
    for (int p = 0; p < NH2; ++p)
#pragma unroll
      for (int a = 0; a < N_IN; ++a) Rm[p][a] = d1[p] * Q1[p][a];

    float g2v[NL];
#pragma unroll
    for (int j = 0; j < NL; ++j) {
      float q0 = 0.f, q1 = 0.f, q2 = 0.f;
#pragma unroll
      for (int p = 0; p < NH2; ++p) {
        q0 = fmaf(W2r[j][p], Rm[p][0], q0);
        q1 = fmaf(W2r[j][p], Rm[p][1], q1);
        q2 = fmaf(W2r[j][p], Rm[p][2], q2);
      }
      g2v[j] = fmaf(q1, q1, fmaf(q0, q0, -(q2 * q2)));
    }

    // ---- psi[j] = d2[j] * (W2[j,:] . v) + s2[j] * g2[j] ----
    float psi[NL];
#pragma unroll
    for (int j = 0; j < NL; ++j) {
      float t = 0.f;
#pragma unroll
      for (int p = 0; p < NH2; ++p) t = fmaf(W2r[j][p], vt[p], t);
      psi[j] = fmaf(d2[j], t, s2[j] * g2v[j]);
    }

    // ---- assemble output ----
    float f = sinf(xv0 + xv1 - xv2);
    float num = f - fmaf(psi[1], lmr[0], fmaf(psi[2], lmr[1], psi[3] * lmr[2]));
    float w0 = num / psi[0];
    float acc = fmaf(h2[1], lmr[0], fmaf(h2[2], lmr[1], h2[3] * lmr[2]));
    out[i] = fmaf(w0, h2[0], acc);
  }
}

extern "C" void kernel_launch(void* const* d_in, const int* in_sizes, int n_in,
                              void* d_out, int out_size, void* d_ws, size_t ws_size,
                              hipStream_t stream) {
  (void)n_in; (void)d_ws; (void)ws_size; (void)in_sizes;
  const float* x  = (const float*)d_in[0];
  const float* W0 = (const float*)d_in[1];
  const float* b0 = (const float*)d_in[2];
  const float* W1 = (const float*)d_in[3];
  const float* b1 = (const float*)d_in[4];
  const float* W2 = (const float*)d_in[5];
  const float* b2 = (const float*)d_in[6];
  const float* lm = (const float*)d_in[7];
  float* out = (float*)d_out;
  const int B = out_size;  // one scalar per sample

  const int threads = 256;                       // 8 wave32 per block
  int blocks = (B + threads - 1) / threads;
  if (blocks > 1024) blocks = 1024;              // grid-stride, ~4 samples/thread
  pinn_fused_kernel<<<blocks, threads, 0, stream>>>(x, W0, b0, W1, b1, W2, b2, lm, out, B);
}